// AdaptiveSudokuLoss_11227044512348
// MI455X (gfx1250) — compile-verified
//
#include <hip/hip_runtime.h>

// ---------------------------------------------------------------------------
// AdaptiveSudokuLoss for MI455X (gfx1250, wave32).
//
// Bandwidth-bound: 191 MB logits @ 23.3 TB/s ~ 8 us floor. Strategy:
//  - 1 wave per grid (grid-stride). Phase A: each lane does full softmax for
//    ~3 cells in registers (CE + entropy accumulated per-lane), stores probs
//    column-major (digit-major, stride 90) into this wave's LDS slice.
//  - Phase B: 27 constraint sums = C(27x81 0/1) x P(81x9) done on the matrix
//    pipe with V_WMMA_F32_16X16X4_F32 (exact f32). A-operand synthesized from
//    per-lane 84-bit constraint masks; B-operand = one ds_load_b64 per k-step.
//    Padded rows/cols masked out using the documented C/D VGPR layout.
//  - Deterministic reduction: wave shuffle -> LDS -> per-block partials in
//    d_ws -> single-block finalize kernel (no float atomics).
// ---------------------------------------------------------------------------

typedef __attribute__((ext_vector_type(2))) float v2f;
typedef __attribute__((ext_vector_type(8))) float v8f;

#define NBLK          1024
#define TPB           256
#define WAVES_PER_BLK 8
#define KSTRIDE       90                  // padded k (cells) stride per digit col
#define LDS_PER_WAVE  (16 * KSTRIDE)      // 16 digit cols x 90 = 1440 floats

__device__ __forceinline__ bool cellInCons(int m, int j) {
  // constraint m (0..26): 0-8 rows, 9-17 cols, 18-26 boxes; cell j in 0..83
  if (m < 9)  return (j / 9) == m;
  if (m < 18) return (j % 9) == (m - 9);
  if (m < 27) { int b = m - 18; return ((j / 27) == (b / 3)) && (((j % 9) / 3) == (b % 3)); }
  return false;
}

__global__ __launch_bounds__(TPB)
void sudoku_main(const float* __restrict__ outs, const int* __restrict__ tgts,
                 float* __restrict__ part, int B) {
  __shared__ __align__(16) float ldsP[WAVES_PER_BLK][LDS_PER_WAVE];
  __shared__ float ldsRed[WAVES_PER_BLK][4];

  const int tid    = threadIdx.x;
  const int lane   = tid & 31;
  const int wv     = tid >> 5;
  const int gwave  = blockIdx.x * WAVES_PER_BLK + wv;
  const int nwaves = gridDim.x * WAVES_PER_BLK;

  float* P = ldsP[wv];

  // One-time clear of this wave's slice: pad digits 9..15 and cells 81..89
  // stay zero forever (phase A rewrites only digit<=8, cell<=80 each grid).
  for (int i = lane; i < LDS_PER_WAVE; i += 32) P[i] = 0.0f;

  // Per-lane constraint bitmasks: A[m][j] for m = T*16 + (lane&15), j=0..83.
  unsigned m0w0 = 0, m0w1 = 0, m0w2 = 0, m1w0 = 0, m1w1 = 0, m1w2 = 0;
  {
    const int mlo = lane & 15;
    #pragma unroll 1
    for (int j = 0; j < 84; ++j) {
      unsigned bit = 1u << (j & 31);
      bool c0 = cellInCons(mlo, j);
      bool c1 = cellInCons(16 + mlo, j);
      if (j < 32)      { if (c0) m0w0 |= bit; if (c1) m1w0 |= bit; }
      else if (j < 64) { if (c0) m0w1 |= bit; if (c1) m1w1 |= bit; }
      else             { if (c0) m0w2 |= bit; if (c1) m1w2 |= bit; }
    }
  }

  const int delta = (lane < 16) ? 0 : 2;   // A/B VGPR0 holds K=0|2, VGPR1 K=1|3
  const int ncol  = lane & 15;             // digit column this lane owns in B/D

  float accCE = 0.f, accENT = 0.f, accCNT = 0.f, accSQ = 0.f;

  const int iters = (B + nwaves - 1) / nwaves;
  for (int it = 0; it < iters; ++it) {
    const int g = gwave + it * nwaves;
    const bool active = (g < B);           // wave-uniform

    if (active) {
      const int gn = g + nwaves;
      if (gn < B) __builtin_prefetch(outs + (size_t)gn * 729, 0, 0);

      // ---- Phase A: per-cell stable softmax, CE + entropy, probs -> LDS ----
      #pragma unroll
      for (int r = 0; r < 3; ++r) {
        const int cell = lane + 32 * r;
        if (cell < 81) {
          const float* xp = outs + ((size_t)g * 81 + cell) * 9;
          float x[9];
          #pragma unroll
          for (int d = 0; d < 9; ++d) x[d] = xp[d];
          const int t = tgts[(size_t)g * 81 + cell];

          float mx = x[0];
          #pragma unroll
          for (int d = 1; d < 9; ++d) mx = fmaxf(mx, x[d]);
          float e[9], se = 0.f;
          #pragma unroll
          for (int d = 0; d < 9; ++d) { e[d] = __expf(x[d] - mx); se += e[d]; }
          const float inv = __frcp_rn(se);
          const float lse = mx + __logf(se);

          float dot = 0.f, xt = 0.f;
          #pragma unroll
          for (int d = 0; d < 9; ++d) {
            const float p = e[d] * inv;
            dot += p * x[d];
            xt  += (d == t) ? x[d] : 0.f;
            P[d * KSTRIDE + cell] = p;      // column-major: (digit, cell)
          }
          accCE += (lse - xt);              // -log p_target
          const float mk = (t != -1) ? 1.f : 0.f;
          accENT += mk * (lse - dot);       // entropy = lse - sum p*x
          accCNT += mk;
        }
      }
    }
    __syncthreads();  // LDS fence: probs visible/ordered before WMMA B-loads

    if (active) {
      // ---- Phase B: S = C x P via f32 WMMA, accumulate (S-1)^2 masked ----
      v8f acc0 = {0.f,0.f,0.f,0.f,0.f,0.f,0.f,0.f};
      v8f acc1 = {0.f,0.f,0.f,0.f,0.f,0.f,0.f,0.f};
      #pragma unroll
      for (int s = 0; s < 21; ++s) {
        const int k0 = 4 * s;
        // B[k][n] for k = k0+delta, k0+delta+1 (one b64: stride-90 col-major)
        v2f b = *(const v2f*)(&P[ncol * KSTRIDE + k0 + delta]);
        // A[m][k] bits: word index is compile-time (pair never crosses words)
        const unsigned w0 = (s < 8) ? m0w0 : ((s < 16) ? m0w1 : m0w2);
        const unsigned w1 = (s < 8) ? m1w0 : ((s < 16) ? m1w1 : m1w2);
        const int sh = (k0 & 31) + delta;
        const unsigned t0 = w0 >> sh;
        const unsigned t1 = w1 >> sh;
        v2f a0; a0.x = (float)(t0 & 1u); a0.y = (float)((t0 >> 1) & 1u);
        v2f a1; a1.x = (float)(t1 & 1u); a1.y = (float)((t1 >> 1) & 1u);
        acc0 = __builtin_amdgcn_wmma_f32_16x16x4_f32(false, a0, false, b,
                                                     (short)0, acc0, false, false);
        acc1 = __builtin_amdgcn_wmma_f32_16x16x4_f32(false, a1, false, b,
                                                     (short)0, acc1, false, false);
      }
      // D layout: vgpr j -> M = j + 8*(lane>=16), N = lane&15
      const int mrowBase = (lane >> 4) * 8;
      const float digOK = (ncol <= 8) ? 1.f : 0.f;
      #pragma unroll
      for (int j = 0; j < 8; ++j) {
        const int c0 = j + mrowBase;        // tile0 constraint id
        const int c1 = 16 + j + mrowBase;   // tile1 constraint id
        const float d0 = acc0[j] - 1.0f;
        const float d1 = acc1[j] - 1.0f;
        accSQ += digOK * ((c0 <= 26) ? d0 * d0 : 0.f);
        accSQ += digOK * ((c1 <= 26) ? d1 * d1 : 0.f);
      }
    }
    __syncthreads();  // before next grid's phase A rewrites LDS
  }

  // ---- deterministic reduction: wave shuffle -> LDS -> block partial ----
  #pragma unroll
  for (int off = 16; off > 0; off >>= 1) {
    accCE  += __shfl_xor(accCE,  off, 32);
    accENT += __shfl_xor(accENT, off, 32);
    accCNT += __shfl_xor(accCNT, off, 32);
    accSQ  += __shfl_xor(accSQ,  off, 32);
  }
  if (lane == 0) {
    ldsRed[wv][0] = accCE; ldsRed[wv][1] = accENT;
    ldsRed[wv][2] = accCNT; ldsRed[wv][3] = accSQ;
  }
  __syncthreads();
  if (tid == 0) {
    float s0 = 0.f, s1 = 0.f, s2 = 0.f, s3 = 0.f;
    for (int w = 0; w < WAVES_PER_BLK; ++w) {
      s0 += ldsRed[w][0]; s1 += ldsRed[w][1];
      s2 += ldsRed[w][2]; s3 += ldsRed[w][3];
    }
    float* o = part + (size_t)blockIdx.x * 4;
    o[0] = s0; o[1] = s1; o[2] = s2; o[3] = s3;
  }
}

__global__ __launch_bounds__(256)
void sudoku_reduce(const float* __restrict__ part, float* __restrict__ out,
                   int nblk, int B) {
  __shared__ float sm[256][4];
  float s0 = 0.f, s1 = 0.f, s2 = 0.f, s3 = 0.f;
  for (int b = threadIdx.x; b < nblk; b += 256) {
    s0 += part[b * 4 + 0]; s1 += part[b * 4 + 1];
    s2 += part[b * 4 + 2]; s3 += part[b * 4 + 3];
  }
  sm[threadIdx.x][0] = s0; sm[threadIdx.x][1] = s1;
  sm[threadIdx.x][2] = s2; sm[threadIdx.x][3] = s3;
  __syncthreads();
  for (int st = 128; st > 0; st >>= 1) {
    if (threadIdx.x < st) {
      #pragma unroll
      for (int c = 0; c < 4; ++c) sm[threadIdx.x][c] += sm[threadIdx.x + st][c];
    }
    __syncthreads();
  }
  if (threadIdx.x == 0) {
    const float ce = sm[0][0], ent = sm[0][1], cnt = sm[0][2], sq = sm[0][3];
    const float fb = (float)B;
    // ce mean over 81B cells; constraint = sum_sq/(243B); conf = ent/(cnt+eps)
    const float loss = ce / (81.0f * fb)
                     + 0.5f * (sq / (243.0f * fb))
                     + 0.1f * (ent / (cnt + 1e-8f));
    out[0] = loss;
  }
}

extern "C" void kernel_launch(void* const* d_in, const int* in_sizes, int n_in,
                              void* d_out, int out_size, void* d_ws, size_t ws_size,
                              hipStream_t stream) {
  const float* outs = (const float*)d_in[0];
  const int*   tgts = (const int*)d_in[1];
  const int B = in_sizes[0] / 729;        // 81 cells * 9 logits
  float* part = (float*)d_ws;             // NBLK * 4 floats = 16 KB

  sudoku_main<<<NBLK, TPB, 0, stream>>>(outs, tgts, part, B);
  sudoku_reduce<<<1, 256, 0, stream>>>(part, (float*)d_out, NBLK, B);
}